// GCN_NEW_52020643889347
// MI455X (gfx1250) — compile-verified
//
#include <hip/hip_runtime.h>
#include <hip/hip_bf16.h>

// ---------------------------------------------------------------------------
// GCN forward for MI455X (gfx1250). Memory/atomic-bound workload; GEMMs use
// the exact-f32 tensor path V_WMMA_F32_16X16X4_F32 (wave32, 16x16 tile/wave).
// ---------------------------------------------------------------------------

typedef float v2f __attribute__((ext_vector_type(2)));
typedef float v8f __attribute__((ext_vector_type(8)));

#define NNODES 50000
#define NFIN   128
#define NH     96
#define NFF    256
#define NG     128
#define BN_EPS 1e-5f

static __device__ __forceinline__ void atomic_add_f32(float* p, float v) {
    __hip_atomic_fetch_add(p, v, __ATOMIC_RELAXED, __HIP_MEMORY_SCOPE_AGENT);
}

// ---------------------------------------------------------------------------
// C[M,Nc] = A[M,K] @ B[K,Nc] (+ bias[Nc] if bias != null)
// One wave32 per 16x16 output tile; K-loop of V_WMMA_F32_16X16X4_F32.
// Requires M%16==0, Nc%16==0, K%4==0 (true for all shapes here).
// A-operand layout: lane L (row M = L&15) holds K = v + 2*(L>>4) in VGPR v.
// B-operand layout: lane L (col N = L&15) holds K = v + 2*(L>>4) in VGPR v.
// D layout: VGPR r -> row r + 8*(L>>4), col L&15.
// ---------------------------------------------------------------------------
__global__ __launch_bounds__(32)
void gemm_wmma_f32(const float* __restrict__ A, const float* __restrict__ B,
                   const float* __restrict__ bias, float* __restrict__ C,
                   int K, int Nc) {
    const int tileN = blockIdx.x;
    const int tileM = blockIdx.y;
    const int lane  = threadIdx.x;
    const int half  = lane >> 4;
    const int l16   = lane & 15;

    const int arow = tileM * 16 + l16;   // A row for this lane
    const int bcol = tileN * 16 + l16;   // B/C column for this lane

    const float* __restrict__ Ap = A + (size_t)arow * K;
    v8f acc = {};

    for (int k0 = 0; k0 < K; k0 += 4) {
        const int ka = k0 + 2 * half;
        v2f a, b;
        a[0] = Ap[ka];
        a[1] = Ap[ka + 1];
        b[0] = B[(size_t)ka * Nc + bcol];
        b[1] = B[(size_t)(ka + 1) * Nc + bcol];
        acc = __builtin_amdgcn_wmma_f32_16x16x4_f32(
            /*neg_a=*/false, a, /*neg_b=*/false, b,
            /*c_mod=*/(short)0, acc, /*reuse_a=*/false, /*reuse_b=*/false);
    }

    const float bi = bias ? bias[bcol] : 0.0f;
#pragma unroll
    for (int r = 0; r < 8; ++r) {
        const int orow = tileM * 16 + r + 8 * half;
        C[(size_t)orow * Nc + bcol] = acc[r] + bi;
    }
}

// ---------------------------------------------------------------------------
// Small elementwise / reduction kernels
// ---------------------------------------------------------------------------
__global__ void fill_f32(float* __restrict__ p, float v, int n) {
    int i = blockIdx.x * blockDim.x + threadIdx.x;
    if (i < n) p[i] = v;
}

__global__ void deg_count(const int* __restrict__ dst, float* __restrict__ deg, int E) {
    int e = blockIdx.x * blockDim.x + threadIdx.x;
    if (e < E) atomic_add_f32(&deg[dst[e]], 1.0f);
}

__global__ void rsqrt_inplace(float* __restrict__ p, int n) {
    int i = blockIdx.x * blockDim.x + threadIdx.x;
    if (i < n) p[i] = rsqrtf(p[i]);
}

// agg[dst] += hw[src] * dis[src]*dis[dst]  (4 features per thread, float4 gather)
__global__ void scatter_edges(const float* __restrict__ hw, const int* __restrict__ src,
                              const int* __restrict__ dst, const float* __restrict__ dis,
                              float* __restrict__ agg, int E, int Hq) {
    int idx = blockIdx.x * blockDim.x + threadIdx.x;
    if (idx >= E * Hq) return;
    int e = idx / Hq;
    int q = idx - e * Hq;
    int s = src[e], d = dst[e];
    float nm = dis[s] * dis[d];
    const float4 v = *((const float4*)(hw + (size_t)s * (Hq * 4)) + q);
    float* ap = agg + (size_t)d * (Hq * 4) + q * 4;
    atomic_add_f32(ap + 0, v.x * nm);
    atomic_add_f32(ap + 1, v.y * nm);
    atomic_add_f32(ap + 2, v.z * nm);
    atomic_add_f32(ap + 3, v.w * nm);
}

// agg = agg + hw * dis[n]^2 + b[c]
__global__ void combine_self(const float* __restrict__ hw, float* __restrict__ agg,
                             const float* __restrict__ dis, const float* __restrict__ b,
                             int Nn, int C) {
    int idx = blockIdx.x * blockDim.x + threadIdx.x;
    if (idx >= Nn * C) return;
    int n = idx / C;
    int c = idx - n * C;
    float dn = dis[n];
    agg[idx] = agg[idx] + hw[idx] * dn * dn + b[c];
}

// one block per channel: mean + rstd over `rows` rows
__global__ void bn_stats(const float* __restrict__ X, int rows, int C,
                         float* __restrict__ mean, float* __restrict__ rstd) {
    __shared__ float s1[256];
    __shared__ float s2[256];
    int c = blockIdx.x;
    float sum = 0.0f, sq = 0.0f;
    for (int r = threadIdx.x; r < rows; r += blockDim.x) {
        float v = X[(size_t)r * C + c];
        sum += v;
        sq += v * v;
    }
    s1[threadIdx.x] = sum;
    s2[threadIdx.x] = sq;
    __syncthreads();
    for (int o = blockDim.x >> 1; o > 0; o >>= 1) {
        if (threadIdx.x < (unsigned)o) {
            s1[threadIdx.x] += s1[threadIdx.x + o];
            s2[threadIdx.x] += s2[threadIdx.x + o];
        }
        __syncthreads();
    }
    if (threadIdx.x == 0) {
        float m   = s1[0] / (float)rows;
        float var = s2[0] / (float)rows - m * m;
        mean[c] = m;
        rstd[c] = rsqrtf(var + BN_EPS);
    }
}

__global__ void bn_relu(const float* __restrict__ X, float* __restrict__ Y,
                        const float* __restrict__ mean, const float* __restrict__ rstd,
                        const float* __restrict__ gamma, const float* __restrict__ beta,
                        int rows, int C) {
    int idx = blockIdx.x * blockDim.x + threadIdx.x;
    if (idx >= rows * C) return;
    int c = idx % C;
    float v = (X[idx] - mean[c]) * rstd[c] * gamma[c] + beta[c];
    Y[idx] = v > 0.0f ? v : 0.0f;
}

__global__ void pool_count(const int* __restrict__ batch, float* __restrict__ cnt, int Nn) {
    int n = blockIdx.x * blockDim.x + threadIdx.x;
    if (n < Nn) atomic_add_f32(&cnt[batch[n]], 1.0f);
}

__global__ void pool_sum(const float* __restrict__ h, const int* __restrict__ batch,
                         float* __restrict__ pooled, int Nn, int C) {
    int idx = blockIdx.x * blockDim.x + threadIdx.x;
    if (idx >= Nn * C) return;
    int n = idx / C;
    int c = idx - n * C;
    atomic_add_f32(&pooled[(size_t)batch[n] * C + c], h[idx]);
}

__global__ void pool_div(float* __restrict__ pooled, const float* __restrict__ cnt, int C, int total) {
    int idx = blockIdx.x * blockDim.x + threadIdx.x;
    if (idx >= total) return;
    pooled[idx] /= fmaxf(cnt[idx / C], 1.0f);
}

// out[g] = relu(dot(z[g,:], w) + b); one wave32 per graph
__global__ __launch_bounds__(32)
void final_linear(const float* __restrict__ z, const float* __restrict__ w,
                  const float* __restrict__ b, float* __restrict__ out, int C) {
    int g = blockIdx.x;
    int lane = threadIdx.x;
    float s = 0.0f;
    for (int c = lane; c < C; c += 32) s += z[(size_t)g * C + c] * w[c];
#pragma unroll
    for (int off = 16; off > 0; off >>= 1) s += __shfl_down(s, off, 32);
    if (lane == 0) {
        float v = s + b[0];
        out[g] = v > 0.0f ? v : 0.0f;
    }
}

// ---------------------------------------------------------------------------
// Host launcher
// ---------------------------------------------------------------------------
extern "C" void kernel_launch(void* const* d_in, const int* in_sizes, int n_in,
                              void* d_out, int out_size, void* d_ws, size_t ws_size,
                              hipStream_t stream) {
    // Input order (lists flattened): x, edge_index, batch,
    // conv_Ws[0..2], conv_bs[0..2], bn_gammas[0..2], bn_betas[0..2],
    // ff_Ws[0..1], ff_bs[0..1], ffbn_gammas[0..1], ffbn_betas[0..1], lin_W, lin_b
    const float* x     = (const float*)d_in[0];
    const int*   eidx  = (const int*)d_in[1];
    const int*   batch = (const int*)d_in[2];
    const float* convW[3] = { (const float*)d_in[3], (const float*)d_in[4], (const float*)d_in[5] };
    const float* convB[3] = { (const float*)d_in[6], (const float*)d_in[7], (const float*)d_in[8] };
    const float* bnG[3]   = { (const float*)d_in[9], (const float*)d_in[10], (const float*)d_in[11] };
    const float* bnB[3]   = { (const float*)d_in[12], (const float*)d_in[13], (const float*)d_in[14] };
    const float* ffW[2]   = { (const float*)d_in[15], (const float*)d_in[16] };
    const float* ffB[2]   = { (const float*)d_in[17], (const float*)d_in[18] };
    const float* ffG[2]   = { (const float*)d_in[19], (const float*)d_in[20] };
    const float* ffBe[2]  = { (const float*)d_in[21], (const float*)d_in[22] };
    const float* linW     = (const float*)d_in[23];
    const float* linB     = (const float*)d_in[24];
    float* out = (float*)d_out;

    const int N = in_sizes[2];          // 50000
    const int E = in_sizes[1] / 2;      // 800000
    const int* src = eidx;
    const int* dst = eidx + E;

    // Workspace layout (floats)
    float* P      = (float*)d_ws;                    // N*NH
    float* Q      = P + (size_t)N * NH;              // N*NH
    float* dis    = Q + (size_t)N * NH;              // N
    float* mean   = dis + N;                         // 256
    float* rstd   = mean + 256;                      // 256
    float* cnt    = rstd + 256;                      // NG
    float* pooled = cnt + NG;                        // NG*NH
    float* z1     = pooled + (size_t)NG * NH;        // NG*NFF
    float* z2     = z1 + (size_t)NG * NFF;           // NG*NFF

    const int TB = 256;
    const size_t nodeFeat = (size_t)N * NH;

    // degree (with self-loop) -> dis = deg^{-1/2}
    fill_f32<<<(N + TB - 1) / TB, TB, 0, stream>>>(dis, 1.0f, N);
    deg_count<<<(E + TB - 1) / TB, TB, 0, stream>>>(dst, dis, E);
    rsqrt_inplace<<<(N + TB - 1) / TB, TB, 0, stream>>>(dis, N);

    // GCN layers (ping-pong P/Q; layer-1 reads x directly)
    const float* hin = x;
    int K = NFIN;
    float* hw  = P;
    float* agg = Q;
    for (int l = 0; l < 3; ++l) {
        gemm_wmma_f32<<<dim3(NH / 16, N / 16), 32, 0, stream>>>(hin, convW[l], nullptr, hw, K, NH);
        hipMemsetAsync(agg, 0, nodeFeat * sizeof(float), stream);
        scatter_edges<<<(int)(((size_t)E * (NH / 4) + TB - 1) / TB), TB, 0, stream>>>(
            hw, src, dst, dis, agg, E, NH / 4);
        combine_self<<<(int)((nodeFeat + TB - 1) / TB), TB, 0, stream>>>(hw, agg, dis, convB[l], N, NH);
        bn_stats<<<NH, 256, 0, stream>>>(agg, N, NH, mean, rstd);
        bn_relu<<<(int)((nodeFeat + TB - 1) / TB), TB, 0, stream>>>(
            agg, hw, mean, rstd, bnG[l], bnB[l], N, NH);
        hin = hw;
        K = NH;
        float* t = hw; hw = agg; agg = t;
    }

    // global mean pool
    hipMemsetAsync(cnt, 0, NG * sizeof(float), stream);
    hipMemsetAsync(pooled, 0, (size_t)NG * NH * sizeof(float), stream);
    pool_count<<<(N + TB - 1) / TB, TB, 0, stream>>>(batch, cnt, N);
    pool_sum<<<(int)((nodeFeat + TB - 1) / TB), TB, 0, stream>>>(hin, batch, pooled, N, NH);
    pool_div<<<(NG * NH + TB - 1) / TB, TB, 0, stream>>>(pooled, cnt, NH, NG * NH);

    // FF layer 1: [G,NH] @ [NH,NFF] + b -> BN -> ReLU (in place)
    gemm_wmma_f32<<<dim3(NFF / 16, NG / 16), 32, 0, stream>>>(pooled, ffW[0], ffB[0], z1, NH, NFF);
    bn_stats<<<NFF, 256, 0, stream>>>(z1, NG, NFF, mean, rstd);
    bn_relu<<<(NG * NFF + TB - 1) / TB, TB, 0, stream>>>(z1, z1, mean, rstd, ffG[0], ffBe[0], NG, NFF);

    // FF layer 2
    gemm_wmma_f32<<<dim3(NFF / 16, NG / 16), 32, 0, stream>>>(z1, ffW[1], ffB[1], z2, NFF, NFF);
    bn_stats<<<NFF, 256, 0, stream>>>(z2, NG, NFF, mean, rstd);
    bn_relu<<<(NG * NFF + TB - 1) / TB, TB, 0, stream>>>(z2, z2, mean, rstd, ffG[1], ffBe[1], NG, NFF);

    // final linear + ReLU
    final_linear<<<NG, 32, 0, stream>>>(z2, linW, linB, out, NFF);
    (void)n_in; (void)out_size; (void)ws_size;
}